// LinearAndGather_936302871102
// MI455X (gfx1250) — compile-verified
//
#include <hip/hip_runtime.h>

// CDNA5 (gfx1250) wave32 WMMA fp32 path.
typedef float v2f __attribute__((ext_vector_type(2)));
typedef float v4f __attribute__((ext_vector_type(4)));
typedef float v8f __attribute__((ext_vector_type(8)));

#define D_DIM 32
#define ITEMS_PER_WAVE 16
#define WAVES_PER_BLOCK 8
#define BLOCK_THREADS (WAVES_PER_BLOCK * 32)
#define ITEMS_PER_BLOCK (WAVES_PER_BLOCK * ITEMS_PER_WAVE)

__global__ __launch_bounds__(BLOCK_THREADS)
void LinearAndGather_936302871102_kernel(
    const float* __restrict__ layer0,   // [N_SRC, 32]  (128 MB: L2-resident, keep RT)
    const float* __restrict__ W,        // [32, 32] row-major (o, i)
    const int*   __restrict__ idx_in,   // [M]          (8 MB: L2-resident, keep RT)
    const int*   __restrict__ idx_out,  // [K]          (streamed once: NT)
    float*       __restrict__ out,      // [K, 32]      (streamed once: NT stores)
    int K, int M, int NSRC)
{
    // Stage W (32x32 fp32 = 4 KB) into LDS once per block, coalesced.
    __shared__ float Wl[D_DIM * D_DIM];
    {
        const v4f* Wv  = (const v4f*)W;
        v4f*       Wlv = (v4f*)Wl;
        Wlv[threadIdx.x] = Wv[threadIdx.x];   // 256 threads * 4 floats = 1024
    }
    __syncthreads();

    const int wave = threadIdx.x >> 5;   // 0..7
    const int lane = threadIdx.x & 31;
    const int m    = lane & 15;          // item slot within tile / W row within tile
    const int hi   = lane >> 4;          // lanes 16-31 carry K-halves 2,3

    const long k  = (long)blockIdx.x * ITEMS_PER_BLOCK + wave * ITEMS_PER_WAVE + m;
    // Clamp instead of branching so EXEC stays all-ones for the WMMAs.
    const long kc = (k < K) ? k : (long)(K - 1);

    // Two dependent random gathers: idx_out -> idx_in -> layer0 row.
    int j = __builtin_nontemporal_load(idx_out + kc);   // one-shot stream: NT
    j = (j < 0) ? 0 : ((j >= M) ? (M - 1) : j);
    int s = idx_in[j];                                  // L2-resident: RT
    s = (s < 0) ? 0 : ((s >= NSRC) ? (NSRC - 1) : s);

    const float* xrow = layer0 + (long)s * D_DIM;

    v8f c0 = {0.f, 0.f, 0.f, 0.f, 0.f, 0.f, 0.f, 0.f};  // out rows 0..15
    v8f c1 = c0;                                        // out rows 16..31

    // A (16x4 f32) layout: lanes 0-15 hold K=0,1 in v[0],v[1]; lanes 16-31 hold K=2,3.
    // B (4x16 f32) layout: VGPR0 = row K=0 (lanes 0-15) / K=2 (lanes 16-31); VGPR1 = K=1 / K=3.
    const float* w0base = &Wl[( 0 + m) * D_DIM + 2 * hi];
    const float* w1base = &Wl[(16 + m) * D_DIM + 2 * hi];
    const float* xbase  = xrow + 2 * hi;

#pragma unroll
    for (int ik = 0; ik < 8; ++ik) {
        v2f a0 = *(const v2f*)(w0base + 4 * ik);   // ds_load_b64
        v2f a1 = *(const v2f*)(w1base + 4 * ik);   // ds_load_b64
        v2f b  = *(const v2f*)(xbase  + 4 * ik);   // global_load_b64 (RT gather, L2 hit)
        c0 = __builtin_amdgcn_wmma_f32_16x16x4_f32(
                 false, a0, false, b, (short)0, c0, false, false);
        c1 = __builtin_amdgcn_wmma_f32_16x16x4_f32(
                 false, a1, false, b, (short)0, c1, false, false);
    }

    if (k < (long)K) {
        // D layout: each lane's 8 accumulator VGPRs are 8 contiguous output floats.
        // tile0 -> out cols (0 + 8*hi), tile1 -> out cols (16 + 8*hi).
        // NT stores: keep the 256 MB output stream from evicting L2-resident layer0.
        float* o0 = out + k * D_DIM + 8 * hi;
        float* o1 = o0 + 16;
        __builtin_nontemporal_store(((v4f*)&c0)[0], (v4f*)(o0 + 0));
        __builtin_nontemporal_store(((v4f*)&c0)[1], (v4f*)(o0 + 4));
        __builtin_nontemporal_store(((v4f*)&c1)[0], (v4f*)(o1 + 0));
        __builtin_nontemporal_store(((v4f*)&c1)[1], (v4f*)(o1 + 4));
    }
}

extern "C" void kernel_launch(void* const* d_in, const int* in_sizes, int n_in,
                              void* d_out, int out_size, void* d_ws, size_t ws_size,
                              hipStream_t stream) {
    const float* layer0  = (const float*)d_in[0];  // N_SRC*32 floats
    const float* W       = (const float*)d_in[1];  // 32*32 floats
    const int*   idx_in  = (const int*)d_in[2];    // M ints
    const int*   idx_out = (const int*)d_in[3];    // K ints
    float*       out     = (float*)d_out;          // K*32 floats

    const int NSRC = in_sizes[0] / (D_DIM);        // layer0 flat = N_SRC*32*1
    const int M    = in_sizes[2];
    const int K    = in_sizes[3];

    const int grid = (K + ITEMS_PER_BLOCK - 1) / ITEMS_PER_BLOCK;
    LinearAndGather_936302871102_kernel<<<grid, BLOCK_THREADS, 0, stream>>>(
        layer0, W, idx_in, idx_out, out, K, M, NSRC);
}